// Net_22720376996275
// MI455X (gfx1250) — compile-verified
//
#include <hip/hip_runtime.h>
#include <math.h>

typedef float v2f __attribute__((ext_vector_type(2)));
typedef float v8f __attribute__((ext_vector_type(8)));

#define DATA_NUM 1024
#define CLS_NUM  32000
#define NTHREADS 1024
#define NWAVES   32
#define NTILES   (CLS_NUM / 16)   // 2000 16-class tiles per row

// ---------------- deterministic block reductions (fixed order) ----------------

__device__ __forceinline__ float blockSum(float v, float* red, float* bc,
                                          int tid, int lane, int wave) {
  #pragma unroll
  for (int off = 16; off > 0; off >>= 1) v += __shfl_down(v, off, 32);
  if (lane == 0) red[wave] = v;
  __syncthreads();
  if (tid == 0) {
    float s = 0.f;
    #pragma unroll
    for (int i = 0; i < NWAVES; ++i) s += red[i];   // fixed serial order
    *bc = s;
  }
  __syncthreads();
  float r = *bc;
  __syncthreads();          // protect red/bc before next reduction reuses them
  return r;
}

__device__ __forceinline__ float blockMax(float v, float* red, float* bc,
                                          int tid, int lane, int wave) {
  #pragma unroll
  for (int off = 16; off > 0; off >>= 1) v = fmaxf(v, __shfl_down(v, off, 32));
  if (lane == 0) red[wave] = v;
  __syncthreads();
  if (tid == 0) {
    float s = -INFINITY;
    #pragma unroll
    for (int i = 0; i < NWAVES; ++i) s = fmaxf(s, red[i]);
    *bc = s;
  }
  __syncthreads();
  float r = *bc;
  __syncthreads();
  return r;
}

// ---------------- per-row AKL kernel: one workgroup per data row ----------------

__global__ __launch_bounds__(NTHREADS)
void akl_row_kernel(const float* __restrict__ h1, const float* __restrict__ e1,
                    const float* __restrict__ h3, const float* __restrict__ e3,
                    float* __restrict__ row_out) {
  extern __shared__ float smem[];
  float* cls = smem;                       // 32000 floats: teacher logits, then p2
  float* red = smem + CLS_NUM;             // 32 wave partials
  float* bc  = smem + CLS_NUM + NWAVES;    // broadcast slot

  const int tid  = threadIdx.x;
  const int lane = tid & 31;
  const int wave = tid >> 5;
  const int row  = blockIdx.x;

  const float h1x = h1[2 * row], h1y = h1[2 * row + 1];
  const float h3x = h3[2 * row], h3y = h3[2 * row + 1];

  const float2* e1v = (const float2*)e1;
  const float2* e3v = (const float2*)e3;

  // ---- B operand: h broadcast to all 16 columns; layout-immune construction.
  // A's K2/K3 lanes are zero, so only B's K0/K1 placement matters; h.y is
  // replicated into both candidate K1 slots.
  v2f b1, b3;
  b1.x = (lane < 16) ? h1x : h1y;
  b1.y = (lane < 16) ? h1y : 0.f;
  b3.x = (lane < 16) ? h3x : h3y;
  b3.y = (lane < 16) ? h3y : 0.f;

  // ---- Pass A: logits via V_WMMA_F32_16X16X4_F32; teacher row -> LDS; maxes.
  float mx1 = -INFINITY, mx2 = -INFINITY;
  for (int t0 = wave; t0 < NTILES + 31; t0 += NWAVES) {
    int tile = (t0 < NTILES) ? t0 : (NTILES - 1);   // wave-uniform clamp (idempotent)
    int cb = tile * 16;

    // A-matrix (16x4 f32): lanes 0-15 hold M=lane, V0=K0, V1=K1; lanes 16-31 = K2/K3 = 0.
    v2f a1 = {0.f, 0.f}, a3 = {0.f, 0.f};
    if (lane < 16) {
      float2 ev1 = e1v[cb + lane];
      float2 ev3 = e3v[cb + lane];
      a1.x = ev1.x; a1.y = ev1.y;
      a3.x = ev3.x; a3.y = ev3.y;
    }

    v8f c1 = {0.f, 0.f, 0.f, 0.f, 0.f, 0.f, 0.f, 0.f};
    v8f c3 = {0.f, 0.f, 0.f, 0.f, 0.f, 0.f, 0.f, 0.f};
    c1 = __builtin_amdgcn_wmma_f32_16x16x4_f32(false, a1, false, b1, (short)0, c1, false, false);
    c3 = __builtin_amdgcn_wmma_f32_16x16x4_f32(false, a3, false, b3, (short)0, c3, false, false);

    #pragma unroll
    for (int k = 0; k < 8; ++k) {
      mx1 = fmaxf(mx1, c1[k]);
      mx2 = fmaxf(mx2, c3[k]);
    }
    // D layout: VGPR v, lanes 0-15 -> M=v ; lanes 16-31 -> M=v+8.
    if (lane == 0) {
      #pragma unroll
      for (int k = 0; k < 8; ++k) cls[cb + k] = c3[k];
    } else if (lane == 16) {
      #pragma unroll
      for (int k = 0; k < 8; ++k) cls[cb + 8 + k] = c3[k];
    }
  }
  const float m1 = blockMax(mx1, red, bc, tid, lane, wave);
  const float m2 = blockMax(mx2, red, bc, tid, lane, wave);

  // ---- Pass B: partition functions.
  float z1 = 0.f, z2 = 0.f;
  for (int j = tid; j < CLS_NUM; j += NTHREADS) {
    float t = cls[j];
    float2 ev = e1v[j];
    float s = fmaf(h1x, ev.x, h1y * ev.y);   // student logit recompute (identical everywhere)
    z2 += __expf(t - m2);
    z1 += __expf(s - m1);
  }
  z1 = blockSum(z1, red, bc, tid, lane, wave);
  z2 = blockSum(z2, red, bc, tid, lane, wave);
  const float l1off = -m1 - __logf(z1);      // lp1 = s + l1off
  const float l2off = -m2 - __logf(z2);      // lp2 = t + l2off

  // ---- Pass C: FKL/RKL; overwrite LDS with teacher probs p2.
  float fkl = 0.f, rkl = 0.f;
  for (int j = tid; j < CLS_NUM; j += NTHREADS) {
    float t = cls[j];
    float2 ev = e1v[j];
    float s = fmaf(h1x, ev.x, h1y * ev.y);
    float lp2 = t + l2off, lp1 = s + l1off;
    float p2 = __expf(lp2), p1 = __expf(lp1);
    fkl += p2 * (lp2 - lp1);
    rkl += p1 * (lp1 - lp2);
    cls[j] = p2;
  }
  fkl = blockSum(fkl, red, bc, tid, lane, wave);
  rkl = blockSum(rkl, red, bc, tid, lane, wave);

  // ---- Pass D: bisection on float-ordered bit keys for the 0.5 cumsum threshold.
  // Find u* = max{u : S(u) < 0.5}, S(u) = sum of p2 with key <= u. Then
  // tail = {j : key(p2_j) <= u*}, exactly tail_mask for distinct probs.
  unsigned lo = 0u, hi = 0x3F800001u;        // keys: 0 .. just above 1.0f
  while (hi - lo > 1u) {
    unsigned mid = lo + ((hi - lo) >> 1);
    float loc = 0.f;
    for (int j = tid; j < CLS_NUM; j += NTHREADS) {
      float p = cls[j];
      if (__float_as_uint(p) <= mid) loc += p;
    }
    float S = blockSum(loc, red, bc, tid, lane, wave);
    if (S < 0.5f) lo = mid; else hi = mid;   // uniform across block
  }

  // ---- Pass E: g_tail / g_head and final per-row AKL.
  float gt = 0.f, gh = 0.f;
  for (int j = tid; j < CLS_NUM; j += NTHREADS) {
    float p2 = cls[j];
    float2 ev = e1v[j];
    float s = fmaf(h1x, ev.x, h1y * ev.y);
    float p1 = __expf(s + l1off);
    float g = fabsf(p2 - p1);
    if (__float_as_uint(p2) <= lo) gt += g; else gh += g;
  }
  gt = blockSum(gt, red, bc, tid, lane, wave);
  gh = blockSum(gh, red, bc, tid, lane, wave);

  if (tid == 0) {
    float den = gh + gt;
    row_out[row] = (gh / den) * fkl + (gt / den) * rkl;
  }
}

// ---------------- deterministic mean over 1024 row values ----------------

__global__ __launch_bounds__(256)
void mean_kernel(const float* __restrict__ rows, float* __restrict__ out) {
  __shared__ float red[8];
  const int tid = threadIdx.x;
  const int lane = tid & 31;
  const int wave = tid >> 5;
  float v = 0.f;
  #pragma unroll
  for (int k = 0; k < 4; ++k) v += rows[tid + k * 256];
  #pragma unroll
  for (int off = 16; off > 0; off >>= 1) v += __shfl_down(v, off, 32);
  if (lane == 0) red[wave] = v;
  __syncthreads();
  if (tid == 0) {
    float s = 0.f;
    #pragma unroll
    for (int i = 0; i < 8; ++i) s += red[i];
    out[0] = s * (1.0f / (float)DATA_NUM);
  }
}

// ---------------- launcher ----------------

extern "C" void kernel_launch(void* const* d_in, const int* in_sizes, int n_in,
                              void* d_out, int out_size, void* d_ws, size_t ws_size,
                              hipStream_t stream) {
  const float* h1 = (const float*)d_in[0];
  const float* e1 = (const float*)d_in[1];
  const float* h3 = (const float*)d_in[2];
  const float* e3 = (const float*)d_in[3];
  // d_in[4] = share_head (0 path)

  float* rows = (float*)d_ws;                       // 1024 floats of scratch
  size_t smem_bytes = (size_t)(CLS_NUM + NWAVES + 1) * sizeof(float);

  akl_row_kernel<<<DATA_NUM, NTHREADS, smem_bytes, stream>>>(h1, e1, h3, e3, rows);
  mean_kernel<<<1, 256, 0, stream>>>(rows, (float*)d_out);
}